// LovaszSoftmaxV1_32744830664867
// MI455X (gfx1250) — compile-verified
//
#include <hip/hip_runtime.h>
#include <math.h>

// Binary Lovasz-Softmax loss, N = 8*1024*1024 pixels.
//   p1 = sigmoid(2x-1); err = |label - p1| (identical for both channels =>
//   one sort serves both). Partition by label, radix-sort each partition's
//   32-bit keys (key = ~bits(err): ascending key == descending err), then
//   merge ranks via binary search and accumulate err * (w_ch0 + w_ch1), *0.5.
//
// CDNA5 specifics:
//   * wave32 ballot-based stable digit ranking (__builtin_amdgcn_ballot_w32)
//   * V_WMMA_F32_16X16X4_F32 powers the 256-wide scans of the radix
//     histograms (integer counts < 2^24 are exact in f32 WMMA, RNE).
//   * All traffic is L2-resident (192 MB >> 140 MB working set).

typedef __attribute__((ext_vector_type(2))) float v2f;
typedef __attribute__((ext_vector_type(8))) float v8f;

#define TILE      2048u            // elements per sort block (256 thr x 8)
#define NBLK_SORT 4096u            // covers 8.39M elements
#define HIST_SIZE (256u * 4096u)   // digit-major transposed histograms (1M)

__device__ __forceinline__ unsigned ballot32(bool p) {
  return __builtin_amdgcn_ballot_w32(p);
}

// ---------------------------------------------------------------------------
// Inclusive 256-element block scan via 8x V_WMMA_F32_16X16X4_F32 (wave 0).
// Values must be non-negative integers < 2^24 held in f32 (exact).
// Stage 1: Y = X * U      (row-wise inclusive prefix; U[k][n] = k<=n)
// Stage 2: W = Ls * Trep  (strict-lower-tri times row totals)
// result(r,c) = Y[r][c] + W[r][c],  element e = r*16 + c.
// Layout assumptions per ISA 7.12.2: A 16x4 f32: vgpr = k&1, lane-half = k>=2,
// lane&15 = row m. B 4x16: vgpr v holds row K=v (lanes 0-15) / K=v+2 (16-31),
// lane&15 = col n. C/D: vgpr v holds row M=v / M=v+8, lane&15 = col n.
// ---------------------------------------------------------------------------
__device__ __forceinline__ float block_scan256_inclusive(float v, float* total)
{
  __shared__ float xs[256];
  __shared__ float ys[256];
  const int tid = threadIdx.x;
  xs[tid] = v;
  __syncthreads();
  if (tid < 32) {                       // wave-uniform branch: EXEC all ones
    const int lane = tid;
    const int hk   = (lane >= 16) ? 2 : 0;   // k-subslot owned by this half
    const int mrow = lane & 15;              // A row
    const int ncol = lane & 15;              // B/C/D column
    v8f acc = {0.f, 0.f, 0.f, 0.f, 0.f, 0.f, 0.f, 0.f};
#pragma unroll
    for (int q = 0; q < 4; ++q) {            // K=16 as 4 chunks of K=4
      const int k0 = 4 * q + hk;
      v2f a; a.x = xs[mrow * 16 + k0];  a.y = xs[mrow * 16 + k0 + 1];
      v2f b; b.x = (k0     <= ncol) ? 1.f : 0.f;
             b.y = (k0 + 1 <= ncol) ? 1.f : 0.f;
      acc = __builtin_amdgcn_wmma_f32_16x16x4_f32(false, a, false, b,
                                                  (short)0, acc, false, false);
    }
    // park Y so row totals (col 15) are fetchable cross-lane
#pragma unroll
    for (int r = 0; r < 8; ++r)
      ys[(r + ((lane >= 16) ? 8 : 0)) * 16 + ncol] = acc[r];
    v8f accw = {0.f, 0.f, 0.f, 0.f, 0.f, 0.f, 0.f, 0.f};
#pragma unroll
    for (int q = 0; q < 4; ++q) {
      const int k0 = 4 * q + hk;
      v2f a; a.x = (k0     < mrow) ? 1.f : 0.f;   // strict lower triangular
             a.y = (k0 + 1 < mrow) ? 1.f : 0.f;
      v2f b; b.x = ys[k0 * 16 + 15];              // T[k] replicated over cols
             b.y = ys[(k0 + 1) * 16 + 15];
      accw = __builtin_amdgcn_wmma_f32_16x16x4_f32(false, a, false, b,
                                                   (short)0, accw, false, false);
    }
#pragma unroll
    for (int r = 0; r < 8; ++r)
      ys[(r + ((lane >= 16) ? 8 : 0)) * 16 + ncol] = acc[r] + accw[r];
  }
  __syncthreads();
  if (total) *total = ys[255];
  return ys[tid];
}

// ---------------------------------------------------------------------------
__global__ void k_init(unsigned* ctr)
{
  if (threadIdx.x < 2) ctr[threadIdx.x] = 0u;
}

// Map + two-sided partition: label==1 keys fill [0, T) upward, label==0 keys
// fill (n-1 .. T] downward. Order inside a partition is irrelevant (sorted
// next), so wave-aggregated global atomics are fine and deterministic-enough:
// equal keys are interchangeable in the final sum.
__global__ void __launch_bounds__(256)
k_map_partition(const float* __restrict__ logits, const int* __restrict__ label,
                unsigned* __restrict__ keys, unsigned* __restrict__ ctr,
                unsigned n)
{
  const unsigned idx  = blockIdx.x * 256u + threadIdx.x;
  const unsigned lane = threadIdx.x & 31u;
  const bool valid = idx < n;
  const float x = valid ? logits[idx] : 0.f;
  const int  lb = valid ? label[idx] : 0;
  const bool one = valid && (lb != 0);
  const float p1  = 1.f / (1.f + __expf(1.f - 2.f * x));   // softmax ch-1
  const float err = fabsf((one ? 1.f : 0.f) - p1);
  const unsigned key = ~__float_as_uint(err);              // asc key = desc err
  const unsigned vmask = ballot32(valid);
  const unsigned m1 = ballot32(one);
  const unsigned m0 = vmask & ~m1;
  unsigned b1 = 0u, b0 = 0u;
  if (lane == 0u && vmask) {            // valid lanes are low-contiguous
    b1 = atomicAdd(&ctr[0], (unsigned)__popc(m1));
    b0 = atomicAdd(&ctr[1], (unsigned)__popc(m0));
  }
  b1 = (unsigned)__shfl((int)b1, 0, 32);
  b0 = (unsigned)__shfl((int)b0, 0, 32);
  const unsigned below = lane ? (0xffffffffu >> (32u - lane)) : 0u;
  if (valid) {
    if (one) keys[b1 + (unsigned)__popc(m1 & below)] = key;
    else     keys[(n - 1u) - (b0 + (unsigned)__popc(m0 & below))] = key;
  }
}

// Per-block 256-bin digit histogram, written digit-major (transposed) so the
// flat exclusive scan directly yields per-(digit,block) scatter bases.
__global__ void __launch_bounds__(256)
k_hist(const unsigned* __restrict__ src, const unsigned* __restrict__ ctr,
       unsigned* __restrict__ histT, int part, int shift, unsigned n)
{
  __shared__ unsigned h[256];
  const unsigned T  = ctr[0];
  const unsigned lo = part ? 0u : T;    // part=1: ones [0,T); part=0: [T,n)
  const unsigned hi = part ? T : n;
  h[threadIdx.x] = 0u;
  __syncthreads();
  const unsigned base = lo + blockIdx.x * TILE;
#pragma unroll
  for (unsigned r = 0; r < 8; ++r) {
    const unsigned idx = base + r * 256u + threadIdx.x;
    if (idx < hi) atomicAdd(&h[(src[idx] >> shift) & 255u], 1u);
  }
  __syncthreads();
  histT[threadIdx.x * NBLK_SORT + blockIdx.x] = h[threadIdx.x];
}

// Scan level 0: per-block (4096-wide) totals of the histogram array.
__global__ void __launch_bounds__(256)
k_scan_reduce(const unsigned* __restrict__ data, float* __restrict__ partials)
{
  __shared__ float red[256];
  const unsigned base = blockIdx.x * 4096u + threadIdx.x * 16u;
  float s = 0.f;
#pragma unroll
  for (unsigned i = 0; i < 16; ++i) s += (float)data[base + i];
  red[threadIdx.x] = s;
  __syncthreads();
  for (int off = 128; off > 0; off >>= 1) {
    if ((int)threadIdx.x < off) red[threadIdx.x] += red[threadIdx.x + off];
    __syncthreads();
  }
  if (threadIdx.x == 0) partials[blockIdx.x] = red[0];
}

// Scan level 1: exclusive-scan the 256 block totals (WMMA scan).
__global__ void __launch_bounds__(256)
k_scan_partials(float* partials)
{
  const float v = partials[threadIdx.x];
  float tot;
  const float incl = block_scan256_inclusive(v, &tot);
  partials[threadIdx.x] = incl - v;
}

// Scan level 2: apply — in-place exclusive scan of histT (WMMA scan).
__global__ void __launch_bounds__(256)
k_scan_apply(unsigned* data, const float* __restrict__ partials)
{
  const unsigned base = blockIdx.x * 4096u + threadIdx.x * 16u;
  float v[16], s = 0.f;
#pragma unroll
  for (unsigned i = 0; i < 16; ++i) { v[i] = (float)data[base + i]; s += v[i]; }
  float tot;
  const float incl = block_scan256_inclusive(s, &tot);
  float run = partials[blockIdx.x] + (incl - s);   // exclusive running base
#pragma unroll
  for (unsigned i = 0; i < 16; ++i) { data[base + i] = (unsigned)run; run += v[i]; }
}

// Stable scatter: 8 tid-ordered rounds of 256; within a round, wave32 ballot
// digit-matching gives lane-ordered ranks; per-wave counts are scanned across
// the 8 waves by thread==digit. Fully stable => multi-pass LSD is correct.
__global__ void __launch_bounds__(256)
k_scatter(const unsigned* __restrict__ src, unsigned* __restrict__ dst,
          const unsigned* __restrict__ ctr, const unsigned* __restrict__ histT,
          int part, int shift, unsigned n)
{
  __shared__ unsigned based[256];
  __shared__ unsigned whist[8][256];
  const unsigned T   = ctr[0];
  const unsigned lo  = part ? 0u : T;
  const unsigned hiE = part ? T : n;
  based[threadIdx.x] = histT[threadIdx.x * NBLK_SORT + blockIdx.x];
  const unsigned lane = threadIdx.x & 31u;
  const unsigned wave = threadIdx.x >> 5;
  const unsigned below = lane ? (0xffffffffu >> (32u - lane)) : 0u;
  for (unsigned r = 0; r < 8; ++r) {
#pragma unroll
    for (unsigned w = 0; w < 8; ++w) whist[w][threadIdx.x] = 0u;
    __syncthreads();
    const unsigned idx = lo + blockIdx.x * TILE + r * 256u + threadIdx.x;
    const bool valid = idx < hiE;
    const unsigned key = valid ? src[idx] : 0u;
    const unsigned dg  = (key >> shift) & 255u;
    const unsigned vmask = ballot32(valid);
    unsigned m = vmask;
#pragma unroll
    for (unsigned b = 0; b < 8; ++b) {             // match_any on 8-bit digit
      const unsigned vb = ballot32((dg >> b) & 1u);
      m &= ((dg >> b) & 1u) ? vb : ~vb;
    }
    m &= vmask;
    const unsigned lrank = (unsigned)__popc(m & below);
    if (valid && lrank == 0u) whist[wave][dg] = (unsigned)__popc(m);
    __syncthreads();
    unsigned excl = 0u;                            // thread == digit
#pragma unroll
    for (unsigned w = 0; w < 8; ++w) {
      const unsigned t = whist[w][threadIdx.x];
      whist[w][threadIdx.x] = excl;
      excl += t;
    }
    const unsigned digitTotal = excl;
    __syncthreads();
    if (valid) dst[lo + based[dg] + whist[wave][dg] + lrank] = key;
    __syncthreads();
    based[threadIdx.x] += digitTotal;              // next read after zero+sync
  }
}

// ---------------------------------------------------------------------------
__device__ __forceinline__ unsigned lower_bound(const unsigned* a, unsigned c,
                                                unsigned key)
{ // first i with a[i] >= key
  unsigned lo = 0, hi = c;
  while (lo < hi) { unsigned mid = (lo + hi) >> 1;
                    if (a[mid] < key) lo = mid + 1; else hi = mid; }
  return lo;
}
__device__ __forceinline__ unsigned upper_bound(const unsigned* a, unsigned c,
                                                unsigned key)
{ // first i with a[i] > key
  unsigned lo = 0, hi = c;
  while (lo < hi) { unsigned mid = (lo + hi) >> 1;
                    if (a[mid] <= key) lo = mid + 1; else hi = mid; }
  return lo;
}
__device__ __forceinline__ float jaccf(float n_c, float i1, float cum)
{ // 1 - (n_c - cum) / (n_c + (i+1) - cum) ; union > 0 always (i1 >= 1)
  return 1.f - (n_c - cum) / (n_c + i1 - cum);
}

// Merge-rank + Lovasz gradient + partial dot products. All ranks/cums < 2^24
// so f32 is exact for the integer parts.
__global__ void __launch_bounds__(256)
k_final(const unsigned* __restrict__ keys, const unsigned* __restrict__ ctr,
        float* __restrict__ blockPartials, unsigned n)
{
  __shared__ float red[256];
  const unsigned T = ctr[0];
  const unsigned idx = blockIdx.x * 256u + threadIdx.x;
  float contrib = 0.f;
  if (idx < n) {
    const bool isOne = idx < T;
    const unsigned key = keys[idx];
    const float err = __uint_as_float(~key);
    const float n1 = (float)T, n0 = (float)(n - T);
    unsigned i; float cum1, oh1;
    if (isOne) {                 // ties: E1 before E0 => strict count of E0
      const unsigned j  = idx;
      const unsigned g0 = lower_bound(keys + T, n - T, key);
      i = j + g0; cum1 = (float)(j + 1u); oh1 = 1.f;
    } else {                     // count E1 with key' <= key (E1 first on tie)
      const unsigned k  = idx - T;
      const unsigned g1 = upper_bound(keys, T, key);
      i = k + g1; cum1 = (float)g1; oh1 = 0.f;
    }
    const float fi1  = (float)(i + 1u);
    const float cum0 = fi1 - cum1;
    const float oh0  = 1.f - oh1;
    const float w1 = jaccf(n1, fi1, cum1) -
                     ((i > 0u) ? jaccf(n1, fi1 - 1.f, cum1 - oh1) : 0.f);
    const float w0 = jaccf(n0, fi1, cum0) -
                     ((i > 0u) ? jaccf(n0, fi1 - 1.f, cum0 - oh0) : 0.f);
    contrib = err * (w1 + w0);
  }
  red[threadIdx.x] = contrib;
  __syncthreads();
  for (int off = 128; off > 0; off >>= 1) {
    if ((int)threadIdx.x < off) red[threadIdx.x] += red[threadIdx.x + off];
    __syncthreads();
  }
  if (threadIdx.x == 0) blockPartials[blockIdx.x] = red[0];
}

// Deterministic fixed-tree reduction of the 32768 block partials; *0.5 = mean
// over the 2 channels.
__global__ void __launch_bounds__(256)
k_reduce(const float* __restrict__ partials, float* __restrict__ out,
         unsigned cnt)
{
  __shared__ float red[256];
  float s = 0.f;
  for (unsigned i = threadIdx.x; i < cnt; i += 256u) s += partials[i];
  red[threadIdx.x] = s;
  __syncthreads();
  for (int off = 128; off > 0; off >>= 1) {
    if ((int)threadIdx.x < off) red[threadIdx.x] += red[threadIdx.x + off];
    __syncthreads();
  }
  if (threadIdx.x == 0) out[0] = 0.5f * red[0];
}

// ---------------------------------------------------------------------------
extern "C" void kernel_launch(void* const* d_in, const int* in_sizes, int n_in,
                              void* d_out, int out_size, void* d_ws,
                              size_t ws_size, hipStream_t stream)
{
  const float* logits = (const float*)d_in[0];
  const int*   label  = (const int*)d_in[1];
  float*       out    = (float*)d_out;
  const unsigned n = (unsigned)in_sizes[0];       // 8,388,608

  // Workspace carve-up (~71.4 MB total, all L2-resident on MI455X):
  char* w = (char*)d_ws;
  unsigned* keysA = (unsigned*)w;  w += (size_t)n * 4u;          // 33.5 MB
  unsigned* keysB = (unsigned*)w;  w += (size_t)n * 4u;          // 33.5 MB
  unsigned* histT = (unsigned*)w;  w += (size_t)HIST_SIZE * 4u;  //  4.0 MB
  float* partials = (float*)w;     w += 256u * 4u;
  float* blockP   = (float*)w;     w += (size_t)((n + 255u) / 256u) * 4u;
  unsigned* ctr   = (unsigned*)w;  w += 64u;

  const unsigned nbMap   = (n + 255u) / 256u;     // 32768
  const unsigned nbScan  = HIST_SIZE / 4096u;     // 256

  k_init<<<1, 64, 0, stream>>>(ctr);
  k_map_partition<<<nbMap, 256, 0, stream>>>(logits, label, keysA, ctr, n);

  unsigned* bufs[2] = {keysA, keysB};
  int cur = 0;
  for (int pass = 0; pass < 4; ++pass) {
    const int shift = pass * 8;
    for (int part = 1; part >= 0; --part) {       // ones [0,T), zeros [T,n)
      k_hist<<<NBLK_SORT, 256, 0, stream>>>(bufs[cur], ctr, histT, part, shift, n);
      k_scan_reduce<<<nbScan, 256, 0, stream>>>(histT, partials);
      k_scan_partials<<<1, 256, 0, stream>>>(partials);
      k_scan_apply<<<nbScan, 256, 0, stream>>>(histT, partials);
      k_scatter<<<NBLK_SORT, 256, 0, stream>>>(bufs[cur], bufs[cur ^ 1], ctr,
                                               histT, part, shift, n);
    }
    cur ^= 1;
  }
  // 4 passes: A->B->A->B->A, sorted result lands back in bufs[0].
  k_final<<<nbMap, 256, 0, stream>>>(bufs[cur], ctr, blockP, n);
  k_reduce<<<1, 256, 0, stream>>>(blockP, out, nbMap);
}